// FocalLoss_7447473291777
// MI455X (gfx1250) — compile-verified
//
#include <hip/hip_runtime.h>

typedef __attribute__((ext_vector_type(2))) float v2f;
typedef __attribute__((ext_vector_type(4))) float v4f;
typedef __attribute__((ext_vector_type(8))) float v8f;

#define MGT 32
#define NCLS 80
#define BLK 256

// Full 32-lane f32 sum: one V_WMMA_F32_16X16X4_F32 (A=ones, B holds lane values)
// folds lane L with lane L^16 (D[m][n] = column-sum of B, independent of m),
// then a 4-step butterfly finishes within each 16-lane half.
__device__ __forceinline__ float wave_sum32(float v) {
  v2f a; a.x = 1.0f; a.y = 1.0f;       // A = ones 16x4
  v2f b; b.x = v;    b.y = 0.0f;       // B rows carry one value per lane
  v8f c = {};
  c = __builtin_amdgcn_wmma_f32_16x16x4_f32(false, a, false, b, (short)0, c, false, false);
  float s = c[0];                      // = v[L%16] + v[(L%16)+16] in every lane
  s += __shfl_xor(s, 1, 32);
  s += __shfl_xor(s, 2, 32);
  s += __shfl_xor(s, 4, 32);
  s += __shfl_xor(s, 8, 32);
  return s;                            // total in all lanes
}

__global__ void fl_init(unsigned long long* packed, int n) {
  int i = blockIdx.x * blockDim.x + threadIdx.x;
  if (i < n) packed[i] = 0ull;
}

// Per-GT argmax over anchors: packed (iou_bits<<32) | (0xFFFFFFFF - anchor)
// so max => larger IoU, ties => smaller anchor index (jnp.argmax semantics).
__global__ void fl_match(const float* __restrict__ anchors,
                         const float* __restrict__ ann1,
                         unsigned long long* __restrict__ packed, int A) {
  __shared__ float a1[MGT * 6];
  __shared__ unsigned long long smax[MGT];
  const int tid = threadIdx.x;
  const int b = blockIdx.y;
  if (tid < MGT * 6) a1[tid] = ann1[b * MGT * 6 + tid];
  if (tid < MGT) smax[tid] = 0ull;
  __syncthreads();
  int a = blockIdx.x * BLK + tid;
  if (a < A) {
    v4f an = ((const v4f*)anchors)[a];
    float areaA = (an.z - an.x) * (an.w - an.y);
    for (int j = 0; j < MGT; ++j) {
      const float* g = &a1[j * 6];
      float areaB = (g[2] - g[0]) * (g[3] - g[1]);
      float iw = fmaxf(fminf(an.z, g[2]) - fmaxf(an.x, g[0]), 0.f);
      float ih = fmaxf(fminf(an.w, g[3]) - fmaxf(an.y, g[1]), 0.f);
      float inter = iw * ih;
      float iou = inter / fmaxf(areaA + areaB - inter, 1e-8f);
      unsigned long long pk = ((unsigned long long)__float_as_uint(iou) << 32)
                            | (unsigned long long)(0xFFFFFFFFu - (unsigned)a);
      atomicMax(&smax[j], pk);
    }
  }
  __syncthreads();
  if (tid < MGT) atomicMax(&packed[(size_t)b * MGT + tid], smax[tid]);
}

__global__ void fl_loss(const float* __restrict__ cls,
                        const float* __restrict__ regr,
                        const float* __restrict__ anchors,
                        const float* __restrict__ ann1,
                        const float* __restrict__ ann2,
                        const unsigned long long* __restrict__ packed,
                        v4f* __restrict__ partials, int A, int nblk) {
  __shared__ float a1[MGT * 6];
  __shared__ float a2[MGT * 6];
  __shared__ int forcedA[MGT];
  __shared__ float red0[BLK], red1[BLK], red2[BLK];
  const int tid = threadIdx.x;
  const int b = blockIdx.y;
  if (tid < MGT * 6) {
    a1[tid] = ann1[b * MGT * 6 + tid];
    a2[tid] = ann2[b * MGT * 6 + tid];
  }
  if (tid < MGT) {
    unsigned long long pk = packed[(size_t)b * MGT + tid];
    forcedA[tid] = (int)(0xFFFFFFFFu - (unsigned)(pk & 0xFFFFFFFFull));
  }
  __syncthreads();

  int a = blockIdx.x * BLK + tid;
  float clsAcc = 0.f, regAcc = 0.f, npAcc = 0.f;
  if (a < A) {
    v4f an = ((const v4f*)anchors)[a];
    float areaA = (an.z - an.x) * (an.w - an.y);
    float best = -1.f; int bi = 0;
    for (int j = 0; j < MGT; ++j) {                 // argmax over 32 GTs (first on tie)
      const float* g = &a1[j * 6];
      float areaB = (g[2] - g[0]) * (g[3] - g[1]);
      float iw = fmaxf(fminf(an.z, g[2]) - fmaxf(an.x, g[0]), 0.f);
      float ih = fmaxf(fminf(an.w, g[3]) - fmaxf(an.y, g[1]), 0.f);
      float inter = iw * ih;
      float iou = inter / fmaxf(areaA + areaB - inter, 1e-8f);
      if (iou > best) { best = iou; bi = j; }
    }
    bool forced = false; int owner = 0;
    for (int j = 0; j < MGT; ++j)                   // last match wins == max j
      if (forcedA[j] == a) { forced = true; owner = j; }
    int bti = forced ? owner : bi;
    bool pos = forced || (best >= 0.5f);
    bool neg = (!forced) && (best < 0.4f);

    if (pos || neg) {                               // ignored anchors contribute 0
      int label = pos ? (int)a1[bti * 6 + 4] : -1;  // -1 => never matches => all-negative path
      const v4f* cp = (const v4f*)(cls + ((size_t)b * A + (size_t)a) * NCLS);
      float l = 0.f;
      #pragma unroll
      for (int i = 0; i < NCLS / 4; ++i) {
        v4f v = __builtin_nontemporal_load(cp + i); // streaming: read-once 256MB array
        #pragma unroll
        for (int k = 0; k < 4; ++k) {
          float p = fminf(fmaxf(v[k], 1e-4f), 1.f - 1e-4f);
          if (4 * i + k == label)
            l += 0.25f * (1.f - p) * (1.f - p) * (-__logf(p));
          else
            l += 0.75f * p * p * (-__logf(1.f - p));
        }
      }
      clsAcc = l;
    }
    if (pos) {
      npAcc = 1.f;
      float aw = an.z - an.x, ah = an.w - an.y;
      float acx = an.x + 0.5f * aw, acy = an.y + 0.5f * ah;
      const float* g = &a1[bti * 6];
      float gw0 = g[2] - g[0], gh0 = g[3] - g[1];
      float gcx = g[0] + 0.5f * gw0, gcy = g[1] + 0.5f * gh0;
      float gw = fmaxf(gw0, 1.f), gh = fmaxf(gh0, 1.f);
      float trk = g[5];
      bool has = false; int fj = 0;                 // first track-id match in frame 2
      for (int j = 0; j < MGT; ++j)
        if (!has && a2[j * 6 + 5] == trk) { has = true; fj = j; }
      float n0 = 0.f, n1 = 0.f, n2 = 0.f, n3 = 0.f;
      if (has) { n0 = a2[fj*6+0]; n1 = a2[fj*6+1]; n2 = a2[fj*6+2]; n3 = a2[fj*6+3]; }
      float nw0 = n2 - n0, nh0 = n3 - n1;
      float ncx = n0 + 0.5f * nw0, ncy = n1 + 0.5f * nh0;
      float nw = fmaxf(nw0, 1.f), nh = fmaxf(nh0, 1.f);
      float t[8];
      t[0] = (gcx - acx) / aw * 10.f; t[1] = (gcy - acy) / ah * 10.f;
      t[2] = __logf(gw / aw) * 5.f;   t[3] = __logf(gh / ah) * 5.f;
      t[4] = (ncx - acx) / aw * 10.f; t[5] = (ncy - acy) / ah * 10.f;
      t[6] = __logf(nw / aw) * 5.f;   t[7] = __logf(nh / ah) * 5.f;
      const v4f* rp = (const v4f*)(regr + ((size_t)b * A + (size_t)a) * 8);
      v4f r0 = rp[0], r1 = rp[1];
      float rv[8] = { r0.x, r0.y, r0.z, r0.w, r1.x, r1.y, r1.z, r1.w };
      float hm = has ? 1.f : 0.f;
      float m[8] = { 1.f, 1.f, 1.f, 1.f, hm, hm, hm, hm };
      float hs = 0.f;
      #pragma unroll
      for (int k = 0; k < 8; ++k) {
        float d = fabsf(t[k] - rv[k]) * m[k];
        hs += (d <= (1.f / 9.f)) ? 4.5f * d * d : (d - (0.5f / 9.f));
      }
      regAcc = hs;
    }
  }
  red0[tid] = clsAcc; red1[tid] = regAcc; red2[tid] = npAcc;
  __syncthreads();
  if (tid < 32) {                                    // wave 0 fully active: EXEC all ones for WMMA
    float c = 0.f, r = 0.f, n = 0.f;
    #pragma unroll
    for (int i = 0; i < BLK / 32; ++i) {
      c += red0[tid + 32 * i]; r += red1[tid + 32 * i]; n += red2[tid + 32 * i];
    }
    c = wave_sum32(c); r = wave_sum32(r); n = wave_sum32(n);
    if (tid == 0) {
      v4f p = { c, r, n, 0.f };
      partials[(size_t)b * nblk + blockIdx.x] = p;
    }
  }
}

__global__ void fl_final(const v4f* __restrict__ partials, float* __restrict__ out,
                         int B, int nblk) {
  __shared__ float sc[32], sr[32];
  const int tid = threadIdx.x;
  const int b = tid >> 5, lane = tid & 31;           // one full wave per sample
  float c = 0.f, r = 0.f, n = 0.f;
  for (int i = lane; i < nblk; i += 32) {
    v4f p = partials[(size_t)b * nblk + i];
    c += p.x; r += p.y; n += p.z;
  }
  c = wave_sum32(c); r = wave_sum32(r); n = wave_sum32(n);
  if (lane == 0) {
    sc[b] = c / fmaxf(n, 1.f);
    sr[b] = r / fmaxf(n * 8.f, 1.f);
  }
  __syncthreads();
  if (tid == 0) {
    float mc = 0.f, mr = 0.f;
    for (int i = 0; i < B; ++i) { mc += sc[i]; mr += sr[i]; }
    out[0] = mc / (float)B;
    out[1] = mr / (float)B;
  }
}

extern "C" void kernel_launch(void* const* d_in, const int* in_sizes, int n_in,
                              void* d_out, int out_size, void* d_ws, size_t ws_size,
                              hipStream_t stream) {
  const float* cls     = (const float*)d_in[0];  // [B,A,80]
  const float* regr    = (const float*)d_in[1];  // [B,A,8]
  const float* anchors = (const float*)d_in[2];  // [1,A,4]
  const float* ann1    = (const float*)d_in[3];  // [B,32,6]
  const float* ann2    = (const float*)d_in[4];  // [B,32,6]
  const int A = in_sizes[2] / 4;
  const int B = (int)((long long)in_sizes[1] / ((long long)A * 8));
  const int nblk = (A + BLK - 1) / BLK;

  unsigned long long* packed = (unsigned long long*)d_ws;           // B*32 u64
  v4f* partials = (v4f*)((char*)d_ws + 4096);                       // B*nblk v4f
  float* out = (float*)d_out;

  fl_init<<<(B * MGT + 255) / 256, 256, 0, stream>>>(packed, B * MGT);
  dim3 grid(nblk, B);
  fl_match<<<grid, BLK, 0, stream>>>(anchors, ann1, packed, A);
  fl_loss<<<grid, BLK, 0, stream>>>(cls, regr, anchors, ann1, ann2, packed, partials, A, nblk);
  fl_final<<<1, B * 32, 0, stream>>>(partials, out, B, nblk);
}